// PHMEmbedding_71648644432298
// MI455X (gfx1250) — compile-verified
//
#include <hip/hip_runtime.h>
#include <cstdint>
#include <cstddef>

// PHM (parameterized-hypercomplex) embedding gather:
//   W = sum_i kron(A[i], S[i]) ; out = W[input]
// Done WITHOUT materializing W: per token t -> (a = t/8000, c = t%8000),
//   out[b*256+d] = sum_{i<4} A[i,a,b] * S[i,c,d]
// Memory-bound (~134 MB total traffic, ~134 MFLOP) => optimize data movement:
//   - async global->LDS b128 staging of the 4 S rows (4 KB) per token
//   - coalesced b128 output stores
// WMMA is not used: K=4 with per-token B operands gives no matrix-core win.

#define NUM_EMB  32000
#define EMB_DIM  1024
#define NCOMP    4
#define RROWS    (NUM_EMB / NCOMP)   /* 8000 */
#define DCOLS    (EMB_DIM / NCOMP)   /* 256  */

__global__ __launch_bounds__(256)
void phm_embed_kernel(const int*   __restrict__ inp,
                      const float* __restrict__ A,     // [4,4,4]  A[i,a,b]
                      const float* __restrict__ S,     // [4,8000,256]
                      float*       __restrict__ out)   // [tokens,1024]
{
    __shared__ __align__(16) float smem[NCOMP * DCOLS];   // 4 KB: S[i, c, :]

    const int bid = blockIdx.x;        // one token per block
    const int tid = threadIdx.x;       // 256 threads = 8 wave32

    const int t = inp[bid];
    const int a = t / RROWS;           // 0..3
    const int c = t - a * RROWS;       // 0..7999

    // ---- Phase 1: async-stage S[i, c, 0:256] for i=0..3 into LDS.
    // Thread tid moves one 16-byte chunk: i = tid>>6, dstart = (tid&63)*4.
    {
        const int    i      = tid >> 6;
        const int    dstart = (tid & 63) << 2;
        const float* gsrc   = S + ((size_t)i * RROWS + (size_t)c) * DCOLS + dstart;
        const unsigned long long gaddr = (unsigned long long)(uintptr_t)gsrc;
        // Low 32 bits of the generic pointer to __shared__ == LDS byte offset.
        const unsigned lds_off = (unsigned)(uintptr_t)(&smem[tid << 2]);
        asm volatile("global_load_async_to_lds_b128 %0, %1, off"
                     :
                     : "v"(lds_off), "v"(gaddr)
                     : "memory");
    }

    // Overlap: fetch the four A coefficients while the async DMA is in flight.
    const int bcol = tid >> 6;               // which b-block (0..3)
    const int lane = tid & 63;               // d/4 within the block
    const float w0 = A[ 0 + (a << 2) + bcol];
    const float w1 = A[16 + (a << 2) + bcol];
    const float w2 = A[32 + (a << 2) + bcol];
    const float w3 = A[48 + (a << 2) + bcol];

    // Drain this wave's async copies, then make LDS visible workgroup-wide.
    asm volatile("s_wait_asynccnt 0" ::: "memory");
    __syncthreads();

    // ---- Phase 2: 4x ds_load_b128 (conflict-free) + fused weighted sum.
    const float4 s0 = *(const float4*)&smem[0 * DCOLS + (lane << 2)];
    const float4 s1 = *(const float4*)&smem[1 * DCOLS + (lane << 2)];
    const float4 s2 = *(const float4*)&smem[2 * DCOLS + (lane << 2)];
    const float4 s3 = *(const float4*)&smem[3 * DCOLS + (lane << 2)];

    float4 r;
    r.x = fmaf(w0, s0.x, fmaf(w1, s1.x, fmaf(w2, s2.x, w3 * s3.x)));
    r.y = fmaf(w0, s0.y, fmaf(w1, s1.y, fmaf(w2, s2.y, w3 * s3.y)));
    r.z = fmaf(w0, s0.z, fmaf(w1, s1.z, fmaf(w2, s2.z, w3 * s3.z)));
    r.w = fmaf(w0, s0.w, fmaf(w1, s1.w, fmaf(w2, s2.w, w3 * s3.w)));

    // Coalesced b128 store: thread tid owns out[row, tid*4 .. tid*4+3].
    *(float4*)(out + (size_t)bid * EMB_DIM + ((size_t)tid << 2)) = r;
}

extern "C" void kernel_launch(void* const* d_in, const int* in_sizes, int n_in,
                              void* d_out, int out_size, void* d_ws, size_t ws_size,
                              hipStream_t stream)
{
    // setup_inputs() dict order: "input" (int32), "A" (f32), "S" (f32)
    const int*   inp = (const int*)  d_in[0];
    const float* A   = (const float*)d_in[1];
    const float* S   = (const float*)d_in[2];
    float*       out = (float*)      d_out;

    const int n_tokens = in_sizes[0];        // 8 * 2048 = 16384

    dim3 grid((unsigned)n_tokens);
    dim3 block(256);
    hipLaunchKernelGGL(phm_embed_kernel, grid, block, 0, stream, inp, A, S, out);
}